// AutomaticBrightnessAndContrast_47562467836653
// MI455X (gfx1250) — compile-verified
//
#include <hip/hip_runtime.h>
#include <stdint.h>

// ---------------------------------------------------------------------------
// AutomaticBrightnessAndContrast for MI455X (gfx1250, wave32).
// Pipeline: [init ws] -> [fused max + dual histogram] -> [params] -> [apply].
// Memory-bound: ~604 MB total traffic => ~26 us at 23.3 TB/s HBM.
// ---------------------------------------------------------------------------

typedef float fvec4 __attribute__((ext_vector_type(4)));
typedef int   v4i   __attribute__((vector_size(16)));   // matches builtin param

#define WS_MAX   0     // 1 uint  : monotonic-mapped float max
#define WS_H255  1     // 256 uint: histogram of image*255
#define WS_H1    257   // 256 uint: histogram of image*1
#define WS_PARAM 513   // 4 float : alpha_eff, beta_eff, hi, adjust_flag
#define WS_WORDS 517

// -------- optional CDNA5 async global->LDS path (ASYNCcnt) -----------------
#if defined(__has_builtin)
#if __has_builtin(__builtin_amdgcn_global_load_async_to_lds_b128)
#define USE_ASYNC_LDS 1
#endif
#endif

#ifdef USE_ASYNC_LDS
typedef __attribute__((address_space(1))) v4i gv4i;   // global int4
typedef __attribute__((address_space(3))) v4i lv4i;   // LDS int4

__device__ __forceinline__ void wait_asynccnt0() {
#if __has_builtin(__builtin_amdgcn_s_wait_asynccnt)
  __builtin_amdgcn_s_wait_asynccnt(0);
#else
  asm volatile("s_wait_asynccnt 0" ::: "memory");
#endif
}
#endif

// -------- float <-> order-preserving uint (for atomicMax on floats) --------
__device__ __forceinline__ unsigned int fmap(float f) {
  unsigned int u = __float_as_uint(f);
  return u ^ (((unsigned int)((int)u >> 31)) | 0x80000000u);
}
__device__ __forceinline__ float funmap(unsigned int u) {
  unsigned int b = (u & 0x80000000u) ? (u ^ 0x80000000u) : ~u;
  return __uint_as_float(b);
}

// -------- exact bin = floor(g / (255/256)), g in [0,255] -------------------
// k * 0.99609375f is exactly representable for 0 <= k <= 256, so we can
// correct a fast multiply-based guess to the true real-arithmetic floor.
__device__ __forceinline__ int bin_of(float g) {
  int b = (int)(g * 1.0039215686f);
  const float bw = 0.99609375f;               // 255/256, exact in f32
  if ((float)b * bw > g)            --b;
  else if ((float)(b + 1) * bw <= g) ++b;
  return b > 255 ? 255 : b;
}

// -------- wave32 ballot-aggregated LDS histogram bump ----------------------
// Fast path: all valid lanes in the wave share one bin -> single ds_add of
// popcount (kills same-address serialization for degenerate distributions).
// Slow path: per-lane ds_add (bins spread -> few bank/addr conflicts).
__device__ __forceinline__ void hist_bump(unsigned int* hist, int bin, bool valid) {
  unsigned long long act = __ballot(valid);
  if (act == 0ull) return;
  int lane   = threadIdx.x & 31;
  int leader = __ffsll(act) - 1;
  int b0     = __shfl(bin, leader, 32);
  unsigned long long same = __ballot(valid && (bin == b0));
  if (same == act) {
    if (lane == leader) atomicAdd(&hist[b0], (unsigned int)__popcll(act));
  } else if (valid) {
    atomicAdd(&hist[bin], 1u);
  }
}

__device__ __forceinline__ void do_pixel(float x, float y, float z,
                                         unsigned int* h255, unsigned int* h1) {
  // scale == 1 branch
  float g1 = 0.299f * x + 0.587f * y + 0.114f * z;
  // scale == 255 branch (reference scales the image first, then weights)
  float x2 = x * 255.0f, y2 = y * 255.0f, z2 = z * 255.0f;
  float g2 = 0.299f * x2 + 0.587f * y2 + 0.114f * z2;
  bool v1 = (g1 >= 0.0f) && (g1 <= 255.0f);
  bool v2 = (g2 >= 0.0f) && (g2 <= 255.0f);
  hist_bump(h1,   v1 ? bin_of(g1) : 0, v1);
  hist_bump(h255, v2 ? bin_of(g2) : 0, v2);
}

// ---------------------------------------------------------------------------
__global__ __launch_bounds__(256) void init_ws_kernel(unsigned int* ws) {
  int i = blockIdx.x * blockDim.x + threadIdx.x;
  if (i < WS_PARAM) ws[i] = 0u;
}

// Fused global-max + dual histogram over the whole image (one 201 MB read).
__global__ __launch_bounds__(256) void hist_kernel(const float* __restrict__ img,
                                                   unsigned int* __restrict__ ws,
                                                   int N /* pixels per plane */) {
  __shared__ unsigned int h255[256];
  __shared__ unsigned int h1[256];
  __shared__ float wmaxs[8];

  const int t = threadIdx.x;
  h255[t] = 0u;
  h1[t]   = 0u;
  __syncthreads();

  const fvec4* __restrict__ p0 = (const fvec4*)img;
  const fvec4* __restrict__ p1 = (const fvec4*)(img + N);
  const fvec4* __restrict__ p2 = (const fvec4*)(img + 2 * N);

  const int groups = N >> 2;
  const int stride = gridDim.x * blockDim.x;
  float vmax = -3.402823466e38f;

  for (int g = blockIdx.x * blockDim.x + t; g < groups; g += stride) {
    fvec4 a = p0[g];           // global_load_b128 x3
    fvec4 b = p1[g];
    fvec4 c = p2[g];

    vmax = fmaxf(vmax,
           fmaxf(fmaxf(fmaxf(a.x, a.y), fmaxf(a.z, a.w)),
           fmaxf(fmaxf(fmaxf(b.x, b.y), fmaxf(b.z, b.w)),
                 fmaxf(fmaxf(c.x, c.y), fmaxf(c.z, c.w)))));

    do_pixel(a.x, b.x, c.x, h255, h1);
    do_pixel(a.y, b.y, c.y, h255, h1);
    do_pixel(a.z, b.z, c.z, h255, h1);
    do_pixel(a.w, b.w, c.w, h255, h1);
  }

  // wave32 max reduce, then block reduce, then one global atomic per block
  for (int off = 16; off > 0; off >>= 1)
    vmax = fmaxf(vmax, __shfl_xor(vmax, off, 32));
  if ((t & 31) == 0) wmaxs[t >> 5] = vmax;
  __syncthreads();
  if (t == 0) {
    float m = wmaxs[0];
    for (int w = 1; w < (int)(blockDim.x >> 5); ++w) m = fmaxf(m, wmaxs[w]);
    atomicMax(&ws[WS_MAX], fmap(m));
  }
  __syncthreads();
  atomicAdd(&ws[WS_H255 + t], h255[t]);
  atomicAdd(&ws[WS_H1 + t],   h1[t]);
}

// Single block: select histogram by is_norm, cumsum, derive affine params.
__global__ __launch_bounds__(256) void params_kernel(unsigned int* __restrict__ ws) {
  __shared__ float acc[256];
  __shared__ unsigned int cnt[2];
  const int t = threadIdx.x;

  float gmax   = funmap(ws[WS_MAX]);
  bool  isnorm = (gmax <= 1.0f);
  float scale  = isnorm ? 255.0f : 1.0f;

  acc[t] = (float)(isnorm ? ws[WS_H255 + t] : ws[WS_H1 + t]);
  if (t < 2) cnt[t] = 0u;
  __syncthreads();

  // Hillis-Steele inclusive scan over 256 bins (all sums <= 2^24, exact f32)
  for (int off = 1; off < 256; off <<= 1) {
    float add = (t >= off) ? acc[t - off] : 0.0f;
    __syncthreads();
    acc[t] += add;
    __syncthreads();
  }

  float maximum = acc[255];
  float clipv   = (maximum / 100.0f) * 0.5f;   // CLIP_HIST_PERCENT = 1.0

  if (acc[t] < clipv)            atomicAdd(&cnt[0], 1u);
  if (acc[t] < maximum - clipv)  atomicAdd(&cnt[1], 1u);
  __syncthreads();

  if (t == 0) {
    int min_gray = (int)cnt[0];
    int max_gray = (int)cnt[1] - 1;
    int span     = max_gray - min_gray;
    if (span < 1) span = 1;
    float alpha = 255.0f / (float)span;
    float beta  = -(float)min_gray * alpha;
    float* pf = (float*)(ws + WS_PARAM);
    pf[0] = alpha / scale;                       // alpha_eff
    pf[1] = beta / scale;                        // beta_eff
    pf[2] = isnorm ? 1.0f : 255.0f;              // hi
    pf[3] = (max_gray > min_gray) ? 1.0f : 0.0f; // adjust?
  }
}

// Streaming affine pass: read 201 MB, write 201 MB. Input staged through LDS
// via async global->LDS (ASYNCcnt) when available; nontemporal b128 stores.
__global__ __launch_bounds__(256) void apply_kernel(const float* __restrict__ img,
                                                    float* __restrict__ out,
                                                    const unsigned int* __restrict__ ws,
                                                    int n4) {
  const float* pf = (const float*)(ws + WS_PARAM);
  const float alpha = pf[0];
  const float beta  = pf[1];
  const float hi    = pf[2];
  const bool  adjust = (pf[3] != 0.0f);

  const fvec4* __restrict__ in4 = (const fvec4*)img;
  fvec4* __restrict__ out4 = (fvec4*)out;
  const int stride = gridDim.x * blockDim.x;

#ifdef USE_ASYNC_LDS
  __shared__ fvec4 stage[256];
  const int t = threadIdx.x;
#endif

  for (int i = blockIdx.x * blockDim.x + threadIdx.x; i < n4; i += stride) {
    fvec4 v;
#ifdef USE_ASYNC_LDS
    // Lane-private global->LDS async copy (no cross-lane, no barrier needed).
    __builtin_amdgcn_global_load_async_to_lds_b128(
        (gv4i*)(in4 + i),
        (lv4i*)(&stage[t]),
        0, 0);
    wait_asynccnt0();
    v = stage[t];                                 // ds_load_b128
#else
    v = __builtin_nontemporal_load(in4 + i);
#endif
    if (adjust) {
      v.x = fminf(fmaxf(v.x * alpha + beta, 0.0f), hi);
      v.y = fminf(fmaxf(v.y * alpha + beta, 0.0f), hi);
      v.z = fminf(fmaxf(v.z * alpha + beta, 0.0f), hi);
      v.w = fminf(fmaxf(v.w * alpha + beta, 0.0f), hi);
    }
    __builtin_nontemporal_store(v, out4 + i);
  }
}

// ---------------------------------------------------------------------------
extern "C" void kernel_launch(void* const* d_in, const int* in_sizes, int n_in,
                              void* d_out, int out_size, void* d_ws, size_t ws_size,
                              hipStream_t stream) {
  const float* img = (const float*)d_in[0];
  float* out = (float*)d_out;
  unsigned int* ws = (unsigned int*)d_ws;

  const int total = in_sizes[0];        // 3 * 4096 * 4096
  const int N     = total / 3;          // pixels per plane
  const int n4    = total / 4;          // fvec4 count (total divisible by 4)

  init_ws_kernel<<<3, 256, 0, stream>>>(ws);
  hist_kernel<<<2048, 256, 0, stream>>>(img, ws, N);
  params_kernel<<<1, 256, 0, stream>>>(ws);
  apply_kernel<<<16384, 256, 0, stream>>>(img, out, ws, n4);
}